// HashEmbedder_59030030516487
// MI455X (gfx1250) — compile-verified
//
#include <hip/hip_runtime.h>

// Instant-NGP multiresolution hash-grid encoding, fused single kernel.
// 4 levels, 2^19-entry tables of float2, trilinear interpolation.
//
// MI455X reasoning: tables total 16 MB -> fully L2-resident (192 MB L2).
// Kernel is L2-gather bound. Strategy: issue all 32 corner gathers per point
// as independent global_load_b64 for deep MLP; keep streaming traffic (x in,
// out) non-temporal so it does not evict the table working set from L2.

typedef float v4f __attribute__((ext_vector_type(4)));
typedef float v2f __attribute__((ext_vector_type(2)));

namespace {
constexpr int      kLevels          = 4;
constexpr int      kRes[kLevels]    = {30, 80, 210, 512};
constexpr unsigned kLog2Entries     = 19;
constexpr unsigned kEntries         = 1u << kLog2Entries;
constexpr unsigned kMask            = kEntries - 1u;
constexpr unsigned kP1              = 11614769u;
constexpr unsigned kP2              = 2654435761u;
}

__global__ __launch_bounds__(256) void hashgrid_fused_kernel(
    const float* __restrict__ x,        // [N,3] in [0,1)
    const v2f*   __restrict__ tables,   // [4, 2^19] float2
    v4f*         __restrict__ out,      // [N,8] f32 viewed as [N,2] float4
    int n)
{
    const int p = blockIdx.x * 256 + threadIdx.x;
    if (p >= n) return;

    // Streamed input: non-temporal so it does not pollute L2 (tables live there).
    const float px = __builtin_nontemporal_load(x + 3 * (size_t)p + 0);
    const float py = __builtin_nontemporal_load(x + 3 * (size_t)p + 1);
    const float pz = __builtin_nontemporal_load(x + 3 * (size_t)p + 2);

    unsigned idx[kLevels][8];
    float    w[kLevels][8];

    // Phase 1: all hash indices + trilinear weights (pure VALU).
#pragma unroll
    for (int li = 0; li < kLevels; ++li) {
        const float r  = (float)kRes[li];
        const float sx = px * r, sy = py * r, sz = pz * r;
        const float bx = floorf(sx), by = floorf(sy), bz = floorf(sz);
        const float tx = sx - bx,   ty = sy - by,   tz = sz - bz;

        const unsigned ix = (unsigned)(int)bx;
        const unsigned iy = (unsigned)(int)by;
        const unsigned iz = (unsigned)(int)bz;

        // hash = (x*1) ^ (y*P1) ^ (z*P2); factor per axis: 2 muls/axis total.
        const unsigned hx[2] = { ix,            ix + 1u       };
        const unsigned hy0   = iy * kP1;
        const unsigned hy[2] = { hy0,           hy0 + kP1     };
        const unsigned hz0   = iz * kP2;
        const unsigned hz[2] = { hz0,           hz0 + kP2     };

        const float wx[2] = { 1.0f - tx, tx };
        const float wy[2] = { 1.0f - ty, ty };
        const float wz[2] = { 1.0f - tz, tz };

#pragma unroll
        for (int c = 0; c < 8; ++c) {
            const int i = (c >> 2) & 1, j = (c >> 1) & 1, k = c & 1;
            idx[li][c] = (hx[i] ^ hy[j] ^ hz[k]) & kMask;
            w[li][c]   = (wx[i] * wy[j]) * wz[k];   // matches reference prod order
        }
    }

    // Phase 2: issue all 32 independent 8-byte gathers back-to-back
    // (global_load_b64 x32 in flight -> hides L2 latency).
    v2f emb[kLevels][8];
#pragma unroll
    for (int li = 0; li < kLevels; ++li) {
        const v2f* __restrict__ tbl = tables + (size_t)li * kEntries;
#pragma unroll
        for (int c = 0; c < 8; ++c) {
            emb[li][c] = tbl[idx[li][c]];
        }
    }

    // Phase 3: trilinear-weighted accumulation.
    float o[8];
#pragma unroll
    for (int li = 0; li < kLevels; ++li) {
        float f0 = 0.0f, f1 = 0.0f;
#pragma unroll
        for (int c = 0; c < 8; ++c) {
            f0 += w[li][c] * emb[li][c].x;
            f1 += w[li][c] * emb[li][c].y;
        }
        o[2 * li + 0] = f0;
        o[2 * li + 1] = f1;
    }

    // Coalesced 32B/lane output, non-temporal (write-once stream).
    v4f lo; lo.x = o[0]; lo.y = o[1]; lo.z = o[2]; lo.w = o[3];
    v4f hi; hi.x = o[4]; hi.y = o[5]; hi.z = o[6]; hi.w = o[7];
    __builtin_nontemporal_store(lo, out + 2 * (size_t)p + 0);
    __builtin_nontemporal_store(hi, out + 2 * (size_t)p + 1);
}

extern "C" void kernel_launch(void* const* d_in, const int* in_sizes, int n_in,
                              void* d_out, int out_size, void* d_ws, size_t ws_size,
                              hipStream_t stream) {
    const float* x      = (const float*)d_in[0];   // [N,3] f32
    const v2f*   tables = (const v2f*)d_in[1];     // [4, 2^19, 2] f32
    v4f*         out    = (v4f*)d_out;             // [N,8] f32

    const int n      = in_sizes[0] / 3;
    const int blocks = (n + 255) / 256;

    hipLaunchKernelGGL(hashgrid_fused_kernel, dim3(blocks), dim3(256), 0, stream,
                       x, tables, out, n);
}